// Attention_27152783246010
// MI455X (gfx1250) — compile-verified
//
#include <hip/hip_runtime.h>

typedef __attribute__((ext_vector_type(2))) float v2f;
typedef __attribute__((ext_vector_type(4))) float v4f;
typedef __attribute__((ext_vector_type(8))) float v8f;

namespace {
constexpr int kB    = 32;
constexpr int kHq   = 32;
constexpr int kHkv  = 8;
constexpr int kG    = 4;     // GQA group size = Hq / Hkv
constexpr int kD    = 128;
constexpr int kBS   = 16;    // paged-KV block size == WMMA tile N
constexpr int kMaxBlocks = 256;
constexpr int kNSplit = 16;                              // flash-decoding splits along T
constexpr int kBlocksPerSplit = kMaxBlocks / kNSplit;    // 16 blocks = 256 tokens/split
constexpr float kScale  = 0.08838834764831845f;          // 1/sqrt(128)
constexpr float kNegInf = -3.0e38f;
}

__device__ __forceinline__ float readlane_f(float v, int srcLane) {
    union { float f; int i; } u;
    u.f = v;
    u.i = __builtin_amdgcn_readlane(u.i, srcLane);
    return u.f;
}

// reductions within each 16-lane half (xor masks stay inside the group)
__device__ __forceinline__ float xormax16(float v) {
#pragma unroll
    for (int m = 8; m >= 1; m >>= 1) v = fmaxf(v, __shfl_xor(v, m, 32));
    return v;
}
__device__ __forceinline__ float xorsum16(float v) {
#pragma unroll
    for (int m = 8; m >= 1; m >>= 1) v += __shfl_xor(v, m, 32);
    return v;
}

// ---------------------------------------------------------------------------
// Kernel 1: per-(b, kv-head, T-split) partial flash attention, one wave32 each.
// S-tile = Q(16x4-padded) x K^T via V_WMMA_F32_16X16X4_F32 chained over D=128.
// Writes unnormalized numerator acc[4][128] plus (m, l) per GQA row.
// ---------------------------------------------------------------------------
__global__ __launch_bounds__(32) void pa_partial_kernel(
    const float* __restrict__ q,            // [B,1,Hq,D]
    const float* __restrict__ k_cache,      // [nblk,16,Hkv,D]
    const float* __restrict__ v_cache,      // [nblk,16,Hkv,D]
    const int*   __restrict__ cache_seqlens,// [B]
    const int*   __restrict__ block_table,  // [B,256]
    float*       __restrict__ acc_buf,      // [B][Hkv][NSplit][G][D]
    float*       __restrict__ ml_buf)       // [B][Hkv][NSplit][G][2] (m,l)
{
    const int wid   = blockIdx.x;
    const int split = wid % kNSplit;
    const int kv    = (wid / kNSplit) % kHkv;
    const int b     = wid / (kNSplit * kHkv);
    const int lane  = threadIdx.x;          // 0..31 (wave32)
    const int n     = lane & 15;            // token-in-tile / M-row id
    const int hi    = lane >> 4;            // which K-pair of the 16x16x4 frag

    const int pidx = (b * kHkv + kv) * kNSplit + split;
    float* accp = acc_buf + (size_t)pidx * (kG * kD);
    float* mlp  = ml_buf  + (size_t)pidx * (kG * 2);

    const int seqlen = cache_seqlens[b];
    const int tok0   = split * kBlocksPerSplit * kBS;
    int ntok = seqlen - tok0;
    if (ntok > kBlocksPerSplit * kBS) ntok = kBlocksPerSplit * kBS;

    if (ntok <= 0) {  // empty split: identity partial (m=-inf, l=0, acc=0)
        if (lane == 0) {
#pragma unroll
            for (int g = 0; g < kG; ++g) { mlp[2 * g] = kNegInf; mlp[2 * g + 1] = 0.0f; }
        }
        const v4f z = {0.0f, 0.0f, 0.0f, 0.0f};
#pragma unroll
        for (int g = 0; g < kG; ++g) *(v4f*)(accp + g * kD + 4 * lane) = z;
        return;
    }
    const int ntiles = (ntok + kBS - 1) / kBS;

    // ---- Q A-fragments, resident for the whole wave (rows g>=4 are zero) ----
    // A layout (16x4 f32): lane = M row (mod 16); VGPR0/1 = K {0,1} (lanes<16)
    // or K {2,3} (lanes>=16).  d = 4*s + 2*hi.
    v2f qa[32];
    {
        const v2f zz = {0.0f, 0.0f};
        if (n < kG) {
            const float* qrow = q + ((size_t)b * kHq + kv * kG + n) * kD;
#pragma unroll
            for (int s = 0; s < 32; ++s) qa[s] = *(const v2f*)(qrow + 4 * s + 2 * hi);
        } else {
#pragma unroll
            for (int s = 0; s < 32; ++s) qa[s] = zz;
        }
    }

    // online-softmax running state (kept wave-uniform via readlane broadcasts)
    float m_u[kG], l_u[kG], acc[kG][4];
#pragma unroll
    for (int g = 0; g < kG; ++g) {
        m_u[g] = kNegInf; l_u[g] = 0.0f;
#pragma unroll
        for (int kk = 0; kk < 4; ++kk) acc[g][kk] = 0.0f;
    }

    const int* bt = block_table + b * kMaxBlocks;
    constexpr int kRowStride = kHkv * kD;   // 1024 floats between tokens

    for (int tile = 0; tile < ntiles; ++tile) {
        const int blk = bt[split * kBlocksPerSplit + tile];  // wave-uniform
        const size_t tile_base = (size_t)blk * (kBS * kRowStride) + (size_t)kv * kD;

        // ---- K B-fragments: lane n owns row t0+n, d = 4s+{0,1}; lane n+16
        // owns d = 4s+{2,3} -> each fragment is one coalesced b64 NT load. ----
        const float* kt = k_cache + tile_base + (size_t)n * kRowStride;
        v2f kb[32];
#pragma unroll
        for (int s = 0; s < 32; ++s)
            kb[s] = __builtin_nontemporal_load((const v2f*)(kt + 4 * s + 2 * hi));

        // ---- S = Q x K^T, 32 chained WMMAs over 4 accumulators ----
        v8f c[4];
#pragma unroll
        for (int i = 0; i < 4; ++i) c[i] = {};
#pragma unroll
        for (int s = 0; s < 32; ++s)
            c[s & 3] = __builtin_amdgcn_wmma_f32_16x16x4_f32(
                false, qa[s], false, kb[s], (short)0, c[s & 3], false, false);
        v8f cs = c[0] + c[1] + c[2] + c[3];
        // S[g, t0+n] lives in component g, lanes 0..15 (rows 4..15 are zero).

        const bool valid = (tok0 + tile * kBS + n) < seqlen;

        float P[kG], alpha[kG];
#pragma unroll
        for (int g = 0; g < kG; ++g) {
            const float Sg   = valid ? cs[g] * kScale : kNegInf;
            const float tmax = readlane_f(xormax16(Sg), 0);       // uniform tile max
            const float mnew = fmaxf(m_u[g], tmax);
            alpha[g] = __expf(m_u[g] - mnew);
            m_u[g]   = mnew;
            P[g]     = __expf(Sg - mnew);                          // 0 for masked
            const float psum = readlane_f(xorsum16(P[g]), 0);
            l_u[g] = l_u[g] * alpha[g] + psum;
#pragma unroll
            for (int kk = 0; kk < 4; ++kk) acc[g][kk] *= alpha[g];
        }

        // ---- acc += P x V ; lane owns d = 4*lane..4*lane+3, p via v_readlane ----
        const float* vt = v_cache + tile_base + 4 * lane;
        v4f vv[kBS];
#pragma unroll
        for (int t = 0; t < kBS; ++t)
            vv[t] = __builtin_nontemporal_load((const v4f*)(vt + (size_t)t * kRowStride));
#pragma unroll
        for (int t = 0; t < kBS; ++t) {
#pragma unroll
            for (int g = 0; g < kG; ++g) {
                const float p = readlane_f(P[g], t);   // SGPR broadcast
                acc[g][0] = fmaf(p, vv[t][0], acc[g][0]);
                acc[g][1] = fmaf(p, vv[t][1], acc[g][1]);
                acc[g][2] = fmaf(p, vv[t][2], acc[g][2]);
                acc[g][3] = fmaf(p, vv[t][3], acc[g][3]);
            }
        }
    }

    if (lane == 0) {
#pragma unroll
        for (int g = 0; g < kG; ++g) { mlp[2 * g] = m_u[g]; mlp[2 * g + 1] = l_u[g]; }
    }
#pragma unroll
    for (int g = 0; g < kG; ++g) {
        const v4f a = {acc[g][0], acc[g][1], acc[g][2], acc[g][3]};
        *(v4f*)(accp + g * kD + 4 * lane) = a;
    }
}

// ---------------------------------------------------------------------------
// Kernel 2: log-sum-exp combine of the NSplit partials -> out[B, Hq, D]
// ---------------------------------------------------------------------------
__global__ __launch_bounds__(128) void pa_combine_kernel(
    const float* __restrict__ acc_buf,
    const float* __restrict__ ml_buf,
    float*       __restrict__ out)
{
    const int bk = blockIdx.x;        // b*Hkv + kv
    const int d  = threadIdx.x;       // 0..127
    const int b  = bk / kHkv;
    const int kv = bk % kHkv;
#pragma unroll
    for (int g = 0; g < kG; ++g) {
        float mstar = kNegInf;
#pragma unroll
        for (int s = 0; s < kNSplit; ++s)
            mstar = fmaxf(mstar, ml_buf[(size_t)(bk * kNSplit + s) * (kG * 2) + 2 * g]);
        float Lsum = 0.0f, num = 0.0f;
#pragma unroll
        for (int s = 0; s < kNSplit; ++s) {
            const size_t pb = (size_t)(bk * kNSplit + s);
            const float ms = ml_buf[pb * (kG * 2) + 2 * g];
            const float ls = ml_buf[pb * (kG * 2) + 2 * g + 1];
            const float w  = __expf(ms - mstar);   // 0 for empty splits
            Lsum = fmaf(w, ls, Lsum);
            num  = fmaf(w, acc_buf[pb * (kG * kD) + g * kD + d], num);
        }
        out[((size_t)b * kHq + kv * kG + g) * kD + d] = num / Lsum;
    }
}

extern "C" void kernel_launch(void* const* d_in, const int* in_sizes, int n_in,
                              void* d_out, int out_size, void* d_ws, size_t ws_size,
                              hipStream_t stream) {
    (void)in_sizes; (void)n_in; (void)out_size; (void)ws_size;
    const float* q   = (const float*)d_in[0];
    const float* kc  = (const float*)d_in[1];
    const float* vc  = (const float*)d_in[2];
    const int*   sl  = (const int*)d_in[3];
    const int*   bt  = (const int*)d_in[4];

    // workspace: acc partials (8 MB) then (m,l) partials (128 KB)
    float* acc_buf = (float*)d_ws;
    float* ml_buf  = (float*)((char*)d_ws +
                     (size_t)kB * kHkv * kNSplit * kG * kD * sizeof(float));

    pa_partial_kernel<<<kB * kHkv * kNSplit, 32, 0, stream>>>(
        q, kc, vc, sl, bt, acc_buf, ml_buf);
    pa_combine_kernel<<<kB * kHkv, 128, 0, stream>>>(
        acc_buf, ml_buf, (float*)d_out);
}